// BloomEmbedding_15745350107437
// MI455X (gfx1250) — compile-verified
//
#include <hip/hip_runtime.h>
#include <stdint.h>

// Bloom-hash embedding gather for MI455X (gfx1250, wave32).
//
// Wave layout: lane -> (h = lane>>3, j = lane&7).
//   h in [0,4): hash/table slab;  j in [0,8): float4 column of the 32-float row.
// Per wave: 4 ids. ids fetched as ONE uniform s_load_b128 (KMcnt), so the
// 4 per-lane global_load_b128 table gathers (LOADcnt) issue back-to-back with
// no intervening loadcnt-0 waits; output written as 512B-coalesced NT b128
// stores so the write-once 419MB output doesn't evict table rows from L2.

typedef __attribute__((ext_vector_type(4))) float f32x4;
typedef __attribute__((ext_vector_type(4))) int   i32x4;

#define TABLE_SIZE 1000000u
#define SUB_DIM    32
#define EMB_DIM    128
#define HASH_SEED  42u
#define IDS_PER_WAVE 4
#define WAVES_PER_BLOCK 8
#define IDS_PER_BLOCK (IDS_PER_WAVE * WAVES_PER_BLOCK)   // 32

__device__ __forceinline__ uint32_t bloom_hash(uint32_t id, uint32_t seed) {
    uint32_t x = id + seed;
    x ^= x >> 16;
    x *= 0x7FEB352Du;
    x ^= x >> 15;
    x *= 0x846CA68Bu;
    x ^= x >> 16;
    return x % TABLE_SIZE;            // lowered to mulhi/mul/sub
}

__global__ void __launch_bounds__(256)
bloom_embedding_kernel(const int* __restrict__ ids,
                       const float* __restrict__ tables,   // [4][1e6][32] f32
                       float* __restrict__ out,            // [N][128] f32
                       int N) {
    const int tid  = threadIdx.x;
    const int lane = tid & 31;
    const int h    = lane >> 3;                     // 0..3
    const int j    = lane & 7;                      // 0..7
    const uint32_t seed = HASH_SEED + (uint32_t)h;

    // Wave-uniform base id index; readfirstlane pins it to an SGPR so the id
    // fetch becomes a scalar (KMcnt) load, decoupled from the LOADcnt gathers.
    int base = (int)(blockIdx.x * WAVES_PER_BLOCK + (tid >> 5)) * IDS_PER_WAVE;
    base = __builtin_amdgcn_readfirstlane(base);

    // Per-lane byte offsets inside the table (all fit in u32: max ~512MB).
    const uint32_t slab_b = (uint32_t)h * (TABLE_SIZE * SUB_DIM * 4u); // h * 128MB
    const uint32_t col_b  = (uint32_t)j * 16u;                         // float4 column

    const unsigned char* tab_b = (const unsigned char*)tables;
    unsigned char*       out_b = (unsigned char*)out;

    if (base + IDS_PER_WAVE <= N) {
        // ---- fast path: branch-free, fully batched ----
        const i32x4 idv = *(const i32x4*)(ids + base);   // uniform -> s_load_b128

        f32x4 v[IDS_PER_WAVE];
#pragma unroll
        for (int u = 0; u < IDS_PER_WAVE; ++u) {
            const uint32_t idx = bloom_hash((uint32_t)idv[u], seed);
            const uint32_t off = slab_b + idx * (SUB_DIM * 4u) + col_b;
            v[u] = *(const f32x4*)(tab_b + (uint64_t)off);   // global_load_b128
        }

#pragma unroll
        for (int u = 0; u < IDS_PER_WAVE; ++u) {
            const uint32_t ooff = (uint32_t)(base + u) * (EMB_DIM * 4u)
                                + (uint32_t)lane * 16u;
            __builtin_nontemporal_store(v[u], (f32x4*)(out_b + (uint64_t)ooff));
        }
    } else {
        // ---- tail path (only if N % 4 != 0 for the last wave) ----
        for (int u = 0; u < IDS_PER_WAVE; ++u) {
            const int n = base + u;
            if (n < N) {
                const uint32_t idx = bloom_hash((uint32_t)ids[n], seed);
                const uint32_t off = slab_b + idx * (SUB_DIM * 4u) + col_b;
                const f32x4 val = *(const f32x4*)(tab_b + (uint64_t)off);
                const uint32_t ooff = (uint32_t)n * (EMB_DIM * 4u)
                                    + (uint32_t)lane * 16u;
                __builtin_nontemporal_store(val, (f32x4*)(out_b + (uint64_t)ooff));
            }
        }
    }
}

extern "C" void kernel_launch(void* const* d_in, const int* in_sizes, int n_in,
                              void* d_out, int out_size, void* d_ws, size_t ws_size,
                              hipStream_t stream) {
    const int*   ids    = (const int*)d_in[0];     // [4096*200] int32
    const float* tables = (const float*)d_in[1];   // [4,1e6,32] f32
    float*       out    = (float*)d_out;           // [N,128] f32
    const int N = in_sizes[0];

    const int grid = (N + IDS_PER_BLOCK - 1) / IDS_PER_BLOCK;  // 25600 for N=819200
    bloom_embedding_kernel<<<dim3(grid), dim3(256), 0, stream>>>(ids, tables, out, N);
}